// RoutingWithSpatialAttention_13692355739666
// MI455X (gfx1250) — compile-verified
//
#include <hip/hip_runtime.h>
#include <math.h>

typedef __attribute__((ext_vector_type(2))) float v2f;
typedef __attribute__((ext_vector_type(8))) float v8f;

// Problem constants
#define B_    32
#define CIN   8
#define COUT  8
#define KK_   4
#define DIN   16
#define DOUT  16
#define SS    32
#define NCH   1024                 // CIN*COUT*DOUT
#define NPIX  1024                 // SS*SS
#define PTS   16384                // DOUT*SS*SS
#define EPSF  1e-5f

// ---------------------------------------------------------------------------
// K1: 3x3 conv (caps [B,16,32,32] -> votes [B,1024,32,32]) via implicit GEMM
// on V_WMMA_F32_16X16X4_F32.  A = Wt tile (16 out-ch x 4 K), B = im2col
// (4 K x 16 pixels), D = [out-ch][pixel]  -> coalesced stores.
// One block per (b, y-row). im2col of the row staged in LDS.
// A fragments cached in registers for the whole N-tile; two accumulator
// chains (both pixel tiles) share each A fragment -> half the global A
// traffic and 2x WMMA ILP.
// ---------------------------------------------------------------------------
#define ASTRIDE 148   // padded LDS row stride (even for b64 align, 148%64=20 -> conflict-free over 16 lanes)

__global__ __launch_bounds__(256) void conv_wmma(const float* __restrict__ caps,
                                                 const float* __restrict__ Wt,
                                                 const float* __restrict__ bt,
                                                 float* __restrict__ votes)
{
    __shared__ float a[SS * ASTRIDE];           // a[x][k], k = i*9 + dy*3 + dx (144)
    const int blk = blockIdx.x;                 // b*32 + y
    const int b = blk >> 5, y = blk & 31;
    const int t = threadIdx.x;

    // Fill im2col (zero-padded 3x3 patches, 16 input channels): 32*144 entries
    for (int idx = t; idx < SS * 144; idx += 256) {
        int x = idx / 144, k = idx - x * 144;
        int i = k / 9, r = k - i * 9;
        int dy = r / 3, dx = r - dy * 3;
        int yy = y + dy - 1, xx = x + dx - 1;
        float val = 0.f;
        if ((unsigned)yy < 32u && (unsigned)xx < 32u)
            val = caps[((b * DIN + i) * SS + yy) * SS + xx];
        a[x * ASTRIDE + k] = val;
    }
    __syncthreads();

    const int wave = t >> 5;
    const int lane = t & 31;
    const int half = lane >> 4;     // 0: K pair {0,1}, 1: K pair {2,3}
    const int l16  = lane & 15;

    const float* brow0 = a + l16 * ASTRIDE + 2 * half;         // pixel tile 0
    const float* brow1 = a + (16 + l16) * ASTRIDE + 2 * half;  // pixel tile 1

    for (int nt = wave; nt < 64; nt += 8) {       // 64 tiles of 16 out-channels
        const int j0 = nt * 16;
        const float* wrow = Wt + (size_t)(j0 + l16) * 144 + 2 * half;

        // Cache all 36 A fragments (16 out-ch x 144 K slice) in registers
        v2f af[36];
        #pragma unroll
        for (int kk = 0; kk < 36; ++kk)
            af[kk] = *(const v2f*)(wrow + 4 * kk);

        v8f acc0 = {0.f, 0.f, 0.f, 0.f, 0.f, 0.f, 0.f, 0.f};
        v8f acc1 = {0.f, 0.f, 0.f, 0.f, 0.f, 0.f, 0.f, 0.f};
        #pragma unroll 6
        for (int kk = 0; kk < 36; ++kk) {         // K = 144 = 36 x 4
            v2f bf0 = *(const v2f*)(brow0 + 4 * kk);
            v2f bf1 = *(const v2f*)(brow1 + 4 * kk);
            acc0 = __builtin_amdgcn_wmma_f32_16x16x4_f32(
                    false, af[kk], false, bf0, (short)0, acc0, false, false);
            acc1 = __builtin_amdgcn_wmma_f32_16x16x4_f32(
                    false, af[kk], false, bf1, (short)0, acc1, false, false);
        }

        // D[m][n]: m = r + 8*half (out-ch within tile), n = l16 (pixel)
        #pragma unroll
        for (int r = 0; r < 8; ++r) {
            int j = j0 + r + 8 * half;
            float bj = bt[j];
            size_t rowbase = (((size_t)b * NCH + j) * SS + y) * SS;
            votes[rowbase + l16]      = acc0[r] + bj;
            votes[rowbase + 16 + l16] = acc1[r] + bj;
        }
    }
}

// ---------------------------------------------------------------------------
// K2: values (einsum omc,bcoduv + bv), pooled max/mean over m (32).
// One block per (b,o); Wv[o],bv[o] staged in LDS.
// votes index: j = c*128 + o*16 + d  ->  offset c*131072 + o*16384 + d*1024 + p
// ---------------------------------------------------------------------------
__global__ __launch_bounds__(256) void values_pool(const float* __restrict__ votes,
                                                   const float* __restrict__ Wv,
                                                   const float* __restrict__ bv,
                                                   float* __restrict__ pmax,
                                                   float* __restrict__ pmean)
{
    __shared__ float wv[256];   // Wv[o][32][8]
    __shared__ float bvv[32];
    const int blk = blockIdx.x;              // b*8 + o
    const int b = blk >> 3, o = blk & 7;
    const int t = threadIdx.x;
    wv[t] = Wv[o * 256 + t];
    if (t < 32) bvv[t] = bv[o * 32 + t];
    __syncthreads();

    const float* vbase = votes + (size_t)b * (NCH * NPIX) + (size_t)o * PTS;
    for (int idx = t; idx < PTS; idx += 256) {
        float vc[8];
        #pragma unroll
        for (int c = 0; c < 8; ++c) vc[c] = vbase[(size_t)c * 131072 + idx];
        float mx = -INFINITY, sm = 0.f;
        #pragma unroll
        for (int m = 0; m < 32; ++m) {
            float s = bvv[m];
            #pragma unroll
            for (int c = 0; c < 8; ++c) s = fmaf(wv[m * 8 + c], vc[c], s);
            mx = fmaxf(mx, s);
            sm += s;
        }
        size_t op = (size_t)blk * PTS + idx;
        pmax[op]  = mx;
        pmean[op] = sm * (1.f / 32.f);
    }
}

// ---------------------------------------------------------------------------
// K0: zero BN accumulators
// ---------------------------------------------------------------------------
__global__ void zero_stats(double* s) { if (threadIdx.x < 16) s[threadIdx.x] = 0.0; }

// ---------------------------------------------------------------------------
// K3: 3x3x3 gate conv over {max,mean} channels (zero pad), store g,
// accumulate per-o sum/sumsq (f64) for BN via global atomics.
// ---------------------------------------------------------------------------
__global__ __launch_bounds__(256) void gate_conv(const float* __restrict__ pmax,
                                                 const float* __restrict__ pmean,
                                                 const float* __restrict__ Ws,
                                                 float* __restrict__ g,
                                                 double* __restrict__ bnstats)
{
    __shared__ float w0[27], w1[27];
    __shared__ double r1[256], r2[256];
    const int blk = blockIdx.x;              // b*8 + o
    const int o = blk & 7;
    const int t = threadIdx.x;
    if (t < 27) { w0[t] = Ws[t]; w1[t] = Ws[27 + t]; }   // ch0 = max, ch1 = mean
    __syncthreads();

    const float* pm = pmax  + (size_t)blk * PTS;
    const float* pa = pmean + (size_t)blk * PTS;
    double s1 = 0.0, s2 = 0.0;
    for (int idx = t; idx < PTS; idx += 256) {
        int d = idx >> 10, p = idx & 1023, y = p >> 5, x = p & 31;
        float acc = 0.f;
        #pragma unroll
        for (int dz = 0; dz < 3; ++dz) {
            int dd = d + dz - 1;
            if ((unsigned)dd >= 16u) continue;
            #pragma unroll
            for (int ry = 0; ry < 3; ++ry) {
                int yy = y + ry - 1;
                if ((unsigned)yy >= 32u) continue;
                #pragma unroll
                for (int rx = 0; rx < 3; ++rx) {
                    int xx = x + rx - 1;
                    if ((unsigned)xx >= 32u) continue;
                    int off = (dd << 10) + (yy << 5) + xx;
                    int wi = (dz * 3 + ry) * 3 + rx;
                    acc = fmaf(w0[wi], pm[off], acc);
                    acc = fmaf(w1[wi], pa[off], acc);
                }
            }
        }
        g[(size_t)blk * PTS + idx] = acc;
        s1 += (double)acc;
        s2 += (double)acc * (double)acc;
    }
    r1[t] = s1; r2[t] = s2;
    __syncthreads();
    for (int off = 128; off > 0; off >>= 1) {
        if (t < off) { r1[t] += r1[t + off]; r2[t] += r2[t + off]; }
        __syncthreads();
    }
    if (t == 0) {
        atomicAdd(&bnstats[o * 2 + 0], r1[0]);
        atomicAdd(&bnstats[o * 2 + 1], r2[0]);
    }
}

// ---------------------------------------------------------------------------
// K3b: finalize BN (mu, 1/sqrt(var+eps)) per o
// ---------------------------------------------------------------------------
__global__ void bn_finalize(const double* __restrict__ bnstats, float* __restrict__ bnout)
{
    int o = threadIdx.x;
    if (o < 8) {
        const double N = (double)B_ * PTS;   // 524288
        double mu  = bnstats[o * 2 + 0] / N;
        double var = bnstats[o * 2 + 1] / N - mu * mu;
        bnout[o * 2 + 0] = (float)mu;
        bnout[o * 2 + 1] = (float)(1.0 / sqrt(var + (double)EPSF));
    }
}

// ---------------------------------------------------------------------------
// K4: gate + agreement routing; writes caps_next into d_out at [o,b,d,y,x]
// ---------------------------------------------------------------------------
__global__ __launch_bounds__(256) void routing(const float* __restrict__ votes,
                                               const float* __restrict__ Wv,
                                               const float* __restrict__ bv,
                                               const float* __restrict__ g,
                                               const float* __restrict__ bnout,
                                               const float* __restrict__ bn_gamma,
                                               const float* __restrict__ bn_beta,
                                               float* __restrict__ out)
{
    __shared__ float wv[256];
    __shared__ float bvv[32];
    const int blk = blockIdx.x;              // b*8 + o
    const int b = blk >> 3, o = blk & 7;
    const int t = threadIdx.x;
    wv[t] = Wv[o * 256 + t];
    if (t < 32) bvv[t] = bv[o * 32 + t];
    __syncthreads();

    const float mu = bnout[o * 2 + 0], rstd = bnout[o * 2 + 1];
    const float gam = bn_gamma[0], bet = bn_beta[0];
    const float* vbase = votes + (size_t)b * (NCH * NPIX) + (size_t)o * PTS;
    const float* gbase = g + (size_t)blk * PTS;
    float* obase = out + ((size_t)o * B_ + b) * PTS;

    for (int idx = t; idx < PTS; idx += 256) {
        float vc[8];
        #pragma unroll
        for (int c = 0; c < 8; ++c) vc[c] = vbase[(size_t)c * 131072 + idx];
        float z = (gbase[idx] - mu) * rstd * gam + bet;
        float factor = 1.f + 1.f / (1.f + __expf(-z));   // values*scale + values

        float xs[8], means[8];
        float xmax = -INFINITY;
        #pragma unroll
        for (int c = 0; c < 8; ++c) {
            float s1 = 0.f, s2 = 0.f;
            #pragma unroll
            for (int k = 0; k < 4; ++k) {
                int m = k * 8 + c;
                float val = bvv[m];
                #pragma unroll
                for (int cc = 0; cc < 8; ++cc) val = fmaf(wv[m * 8 + cc], vc[cc], val);
                val *= factor;
                s1 += val;
                s2 = fmaf(val, val, s2);
            }
            float mean = 0.25f * s1;
            float var  = fmaxf(0.25f * s2 - mean * mean, 0.f);
            float x = -logf(sqrtf(var));               // -log(std); softmax over c
            xs[c] = x; means[c] = mean;
            xmax = fmaxf(xmax, x);
        }
        float esum = 0.f, acc = 0.f;
        #pragma unroll
        for (int c = 0; c < 8; ++c) {
            float e = __expf(xs[c] - xmax);
            esum += e;
            acc = fmaf(e, means[c], acc);
        }
        obase[idx] = acc / esum;
    }
}

// ---------------------------------------------------------------------------
// K5: LayerNorm over [D,32,32] per (o,b), in place on d_out
// ---------------------------------------------------------------------------
__global__ __launch_bounds__(256) void layernorm(float* __restrict__ out,
                                                 const float* __restrict__ ln_g,
                                                 const float* __restrict__ ln_b)
{
    __shared__ double r1[256], r2[256];
    float* base = out + (size_t)blockIdx.x * PTS;     // blockIdx = o*32+b
    const int t = threadIdx.x;
    double s1 = 0.0, s2 = 0.0;
    for (int idx = t; idx < PTS; idx += 256) {
        float v = base[idx];
        s1 += (double)v;
        s2 += (double)v * (double)v;
    }
    r1[t] = s1; r2[t] = s2;
    __syncthreads();
    for (int off = 128; off > 0; off >>= 1) {
        if (t < off) { r1[t] += r1[t + off]; r2[t] += r2[t + off]; }
        __syncthreads();
    }
    double m = r1[0] / (double)PTS;
    double var = r2[0] / (double)PTS - m * m;
    float mf = (float)m;
    float inv = (float)(1.0 / sqrt(var + (double)EPSF));
    for (int idx = t; idx < PTS; idx += 256) {
        float v = base[idx];
        base[idx] = (v - mf) * inv * ln_g[idx] + ln_b[idx];
    }
}

// ---------------------------------------------------------------------------
extern "C" void kernel_launch(void* const* d_in, const int* in_sizes, int n_in,
                              void* d_out, int out_size, void* d_ws, size_t ws_size,
                              hipStream_t stream)
{
    const float* caps     = (const float*)d_in[0];
    const float* Wt       = (const float*)d_in[1];
    const float* bt       = (const float*)d_in[2];
    const float* Wv       = (const float*)d_in[3];
    const float* bv       = (const float*)d_in[4];
    const float* Ws       = (const float*)d_in[5];
    const float* bn_gamma = (const float*)d_in[6];
    const float* bn_beta  = (const float*)d_in[7];
    const float* ln_gamma = (const float*)d_in[8];
    const float* ln_beta  = (const float*)d_in[9];
    float* out = (float*)d_out;

    // Workspace layout (floats)
    float* ws      = (float*)d_ws;
    float* votes   = ws;                               // 33,554,432
    float* pmax    = ws + 33554432;                    //  4,194,304
    float* pmean   = ws + 37748736;                    //  4,194,304
    float* gbuf    = ws + 41943040;                    //  4,194,304
    double* bnst   = (double*)(ws + 46137344);         // 16 doubles
    float* bnout   = ws + 46137344 + 32;               // 16 floats

    conv_wmma  <<<B_ * SS,      256, 0, stream>>>(caps, Wt, bt, votes);
    values_pool<<<B_ * COUT,    256, 0, stream>>>(votes, Wv, bv, pmax, pmean);
    zero_stats <<<1, 32,           0, stream>>>(bnst);
    gate_conv  <<<B_ * COUT,    256, 0, stream>>>(pmax, pmean, Ws, gbuf, bnst);
    bn_finalize<<<1, 8,            0, stream>>>(bnst, bnout);
    routing    <<<B_ * COUT,    256, 0, stream>>>(votes, Wv, bv, gbuf, bnout,
                                                  bn_gamma, bn_beta, out);
    layernorm  <<<COUT * B_,    256, 0, stream>>>(out, ln_gamma, ln_beta);
}